// Autoencoder_85246510891724
// MI455X (gfx1250) — compile-verified
//
#include <hip/hip_runtime.h>
#include <math.h>

#define DT 0.03125f
#define NSTEPS 111

__device__ __forceinline__ float4 lerp4(float4 a, float4 b, float t) {
    float4 r;
    r.x = fmaf(t, b.x - a.x, a.x);
    r.y = fmaf(t, b.y - a.y, a.y);
    r.z = fmaf(t, b.z - a.z, a.z);
    r.w = fmaf(t, b.w - a.w, a.w);
    return r;
}

// Pre-pass: planar [B,4,D,D,D] -> interleaved [B,D,D,D,4] (one float4 per voxel).
// 64MB -> 64MB; HBM-bound and tiny vs. the march (~6us at 23.3 TB/s).
__global__ __launch_bounds__(256) void transpose_vol(const float* __restrict__ vol,
                                                     float4* __restrict__ vol4,
                                                     int nvox) {
    int b = blockIdx.y;
    size_t i = (size_t)blockIdx.x * blockDim.x + threadIdx.x;
    if (i >= (size_t)nvox) return;
    const float* v = vol + (size_t)b * 4u * nvox;
    float4 o;
    o.x = v[i];
    o.y = v[i + (size_t)nvox];
    o.z = v[i + 2u * (size_t)nvox];
    o.w = v[i + 3u * (size_t)nvox];
    vol4[(size_t)b * nvox + i] = o;
}

// One thread per ray. INTER=true samples the interleaved float4 volume
// (global_load_b128 with immediate offsets when DFIX>0); INTER=false gathers
// from the planar reference layout (no workspace needed). DFIX=0 -> runtime D.
template <bool INTER, int DFIX>
__global__ __launch_bounds__(256) void raymarch(const float* __restrict__ camrot,
                                                const float* __restrict__ campos,
                                                const float* __restrict__ focal,
                                                const float* __restrict__ princpt,
                                                const float* __restrict__ pixelcoords,
                                                const float* __restrict__ vol,
                                                const float4* __restrict__ vol4,
                                                float* __restrict__ out,
                                                int H, int W, int Dparam) {
    const int D = (DFIX > 0) ? DFIX : Dparam;

    int w = blockIdx.x * blockDim.x + threadIdx.x;
    int h = blockIdx.y * blockDim.y + threadIdx.y;
    int b = blockIdx.z;
    if (w >= W || h >= H) return;

    const float* R = camrot + b * 9;
    const float cx = campos[b * 3 + 0], cy = campos[b * 3 + 1], cz = campos[b * 3 + 2];
    const float flx = focal[b * 2 + 0], fly = focal[b * 2 + 1];
    const float ppx = princpt[b * 2 + 0], ppy = princpt[b * 2 + 1];

    size_t pidx = ((size_t)b * H + h) * W + w;
    const float pcx = pixelcoords[pidx * 2 + 0];
    const float pcy = pixelcoords[pidx * 2 + 1];

    // pixel -> camera -> world (camrot^T as in einsum('bij,bhwi->bhwj'))
    float dx = (pcx - ppx) / flx;
    float dy = (pcy - ppy) / fly;
    float dz = 1.0f;
    float rx = R[0] * dx + R[3] * dy + R[6] * dz;
    float ry = R[1] * dx + R[4] * dy + R[7] * dz;
    float rz = R[2] * dx + R[5] * dy + R[8] * dz;
    float nrm = sqrtf(rx * rx + ry * ry + rz * rz);
    rx /= nrm; ry /= nrm; rz /= nrm;

    // ray-AABB [-1,1]^3
    float t1x = (-1.0f - cx) / rx, t2x = (1.0f - cx) / rx;
    float t1y = (-1.0f - cy) / ry, t2y = (1.0f - cy) / ry;
    float t1z = (-1.0f - cz) / rz, t2z = (1.0f - cz) / rz;
    float tmin = fmaxf(fminf(t1x, t2x), fmaxf(fminf(t1y, t2y), fminf(t1z, t2z)));
    float tmax = fminf(fmaxf(t1x, t2x), fminf(fmaxf(t1y, t2y), fmaxf(t1z, t2z)));
    float t0 = (tmin < tmax) ? fmaxf(tmin, 0.0f) : 0.0f;

    float px = cx + rx * t0;
    float py = cy + ry * t0;
    float pz = cz + rz * t0;

    const int nvox = D * D * D;
    const float* __restrict__ vb = vol + (size_t)b * 4u * (size_t)nvox;
    const float4* __restrict__ v4 = vol4 + (size_t)b * (size_t)nvox;
    const float nmax = (float)(D - 1);
    const int zstride = D * D;

    // Per-ray voxel delta per step (for speculative prefetch one step ahead).
    const float vstep = DT * 0.5f * nmax;
    const int dix = (int)floorf(rx * vstep + 0.5f);
    const int diy = (int)floorf(ry * vstep + 0.5f);
    const int diz = (int)floorf(rz * vstep + 0.5f);
    const long pdelta = ((long)diz * D + diy) * D + dix;

    float r0 = 0.0f, r1 = 0.0f, r2 = 0.0f, ra = 0.0f;
    bool entered = false;

    for (int s = 0; s < NSTEPS; ++s) {
        bool valid = (px > -1.0f) && (px < 1.0f) &&
                     (py > -1.0f) && (py < 1.0f) &&
                     (pz > -1.0f) && (pz < 1.0f);
        if (valid) {
            entered = true;
            // grid coords, exact op order as reference: ((p+1)*0.5)*(n-1), clipped
            float gx = fminf(fmaxf((px + 1.0f) * 0.5f * nmax, 0.0f), nmax);
            float gy = fminf(fmaxf((py + 1.0f) * 0.5f * nmax, 0.0f), nmax);
            float gz = fminf(fmaxf((pz + 1.0f) * 0.5f * nmax, 0.0f), nmax);
            int ix = min((int)gx, D - 2);
            int iy = min((int)gy, D - 2);
            int iz = min((int)gz, D - 2);
            float fx = gx - (float)ix;
            float fy = gy - (float)iy;
            float fz = gz - (float)iz;

            float4 s000, s001, s010, s011, s100, s101, s110, s111;
            // With DFIX=128 this is shifts/ors and the 8 loads below become a
            // single base address + immediate offsets (y: +2KB, z: +256KB, both
            // inside the 24-bit IOFFSET of GLOBAL_LOAD_B128).
            size_t base = ((size_t)iz * D + iy) * D + ix;
            if (INTER) {
                const float4* p0 = v4 + base;
                // speculative prefetch of next step's neighborhood (global_prefetch_b8)
                __builtin_prefetch((const void*)(p0 + pdelta), 0, 1);
                __builtin_prefetch((const void*)(p0 + pdelta + D), 0, 1);
                s000 = p0[0];
                s001 = p0[1];
                s010 = p0[D];
                s011 = p0[D + 1];
                s100 = p0[zstride];
                s101 = p0[zstride + 1];
                s110 = p0[zstride + D];
                s111 = p0[zstride + D + 1];
            } else {
                auto g = [&](size_t off) {
                    float4 r;
                    r.x = vb[off];
                    r.y = vb[off + (size_t)nvox];
                    r.z = vb[off + 2u * (size_t)nvox];
                    r.w = vb[off + 3u * (size_t)nvox];
                    return r;
                };
                s000 = g(base);
                s001 = g(base + 1);
                s010 = g(base + D);
                s011 = g(base + D + 1);
                s100 = g(base + zstride);
                s101 = g(base + zstride + 1);
                s110 = g(base + zstride + D);
                s111 = g(base + zstride + D + 1);
            }

            float4 cA = lerp4(s000, s001, fx);  // z0,y0
            float4 cB = lerp4(s010, s011, fx);  // z0,y1
            float4 cC = lerp4(s100, s101, fx);  // z1,y0
            float4 cD = lerp4(s110, s111, fx);  // z1,y1
            float4 c0 = lerp4(cA, cB, fy);
            float4 c1 = lerp4(cC, cD, fy);
            float4 smp = lerp4(c0, c1, fz);

            float contrib = fminf(ra + smp.w * DT, 1.0f) - ra;
            r0 += smp.x * contrib;
            r1 += smp.y * contrib;
            r2 += smp.z * contrib;
            ra += contrib;
        } else if (entered) {
            break;  // cube is convex: once exited, never valid again
        }
        px += rx * DT;
        py += ry * DT;
        pz += rz * DT;
    }

    size_t hw = (size_t)h * W + w;
    size_t bo = (size_t)b * 4u * H * W;
    out[bo + 0u * (size_t)H * W + hw] = r0;
    out[bo + 1u * (size_t)H * W + hw] = r1;
    out[bo + 2u * (size_t)H * W + hw] = r2;
    out[bo + 3u * (size_t)H * W + hw] = ra;
}

extern "C" void kernel_launch(void* const* d_in, const int* in_sizes, int n_in,
                              void* d_out, int out_size, void* d_ws, size_t ws_size,
                              hipStream_t stream) {
    const float* camrot = (const float*)d_in[0];
    const float* campos = (const float*)d_in[1];
    const float* focal = (const float*)d_in[2];
    const float* princpt = (const float*)d_in[3];
    const float* pixelcoords = (const float*)d_in[4];
    const float* volume = (const float*)d_in[5];
    float* out = (float*)d_out;

    const int B = in_sizes[1] / 3;                 // campos = B*3
    const int HW = in_sizes[4] / (B * 2);          // pixelcoords = B*H*W*2
    const int Hd = (int)(sqrt((double)HW) + 0.5);  // square image per reference
    const int Wd = Hd;
    const int nvox = in_sizes[5] / (B * 4);        // volume = B*4*D^3
    const int D = (int)(cbrt((double)nvox) + 0.5);

    dim3 block(32, 8, 1);
    dim3 grid((Wd + 31) / 32, (Hd + 7) / 8, B);

    size_t need = (size_t)B * (size_t)nvox * sizeof(float4);
    if (D == 128 && ws_size >= need) {
        // Fast path: interleave volume into d_ws, march with compile-time D=128.
        float4* vol4 = (float4*)d_ws;
        dim3 tblock(256, 1, 1);
        dim3 tgrid((unsigned)((nvox + 255) / 256), (unsigned)B, 1);
        transpose_vol<<<tgrid, tblock, 0, stream>>>(volume, vol4, nvox);
        raymarch<true, 128><<<grid, block, 0, stream>>>(camrot, campos, focal, princpt,
                                                        pixelcoords, volume, vol4, out,
                                                        Hd, Wd, D);
    } else {
        // Generic fallback: planar gathers straight from the input volume.
        raymarch<false, 0><<<grid, block, 0, stream>>>(camrot, campos, focal, princpt,
                                                       pixelcoords, volume,
                                                       (const float4*)nullptr, out,
                                                       Hd, Wd, D);
    }
}